// EfficientDetWithPostProcessing_66718021976316
// MI455X (gfx1250) — compile-verified
//
#include <hip/hip_runtime.h>
#include <hip/hip_bf16.h>
#include <stdint.h>

#define A_TOT   110484
#define NCLS    90
#define KC      512
#define MAXDET  100
#define NFLAT   (NCLS * KC)          /* 46080 */
#define TPB_SEL 1024
#define EPT     45                   /* 46080 / 1024 exactly */
#define IMGF    768.0f
#define THRESH  0.05f
#define IOU_THR 0.5f
#define TR_ANCH 128                  /* anchors per transpose tile */

static_assert(NFLAT == TPB_SEL * EPT, "flat layout must tile exactly");

// ---- TDM availability (probe-verified builtin; arity differs per toolchain)
#if defined(__has_builtin)
#if __has_builtin(__builtin_amdgcn_tensor_load_to_lds) && \
    __has_builtin(__builtin_amdgcn_s_wait_tensorcnt)
#define HAVE_TDM 1
#endif
#endif
#ifndef HAVE_TDM
#define HAVE_TDM 0
#endif

typedef unsigned __attribute__((ext_vector_type(4))) v4u;
typedef int __attribute__((ext_vector_type(4))) v4i;
typedef int __attribute__((ext_vector_type(8))) v8i;

// ---- monotonic float <-> uint key (ascending order preserved) ----
__device__ __forceinline__ unsigned fkey(float f) {
  unsigned b = __float_as_uint(f);
  return b ^ ((b & 0x80000000u) ? 0xFFFFFFFFu : 0x80000000u);
}
__device__ __forceinline__ float unfkey(unsigned k) {
  unsigned b = k ^ ((k & 0x80000000u) ? 0x80000000u : 0xFFFFFFFFu);
  return __uint_as_float(b);
}

// =====================================================================
// Kernel 1: decode + clip boxes  (elementwise, float4 path)
// =====================================================================
__global__ __launch_bounds__(256) void decode_kernel(
    const float* __restrict__ reg, const float* __restrict__ anc,
    float* __restrict__ boxes) {
  int a = blockIdx.x * blockDim.x + threadIdx.x;
  if (a >= A_TOT) return;
  float4 an = ((const float4*)anc)[a];
  float4 d  = ((const float4*)reg)[a];
  float wa = an.z - an.x, ha = an.w - an.y;
  float cxa = an.x + 0.5f * wa, cya = an.y + 0.5f * ha;
  float cx = cxa + d.x * wa, cy = cya + d.y * ha;
  float w = expf(d.z) * wa, h = expf(d.w) * ha;
  float x1 = fminf(fmaxf(cx - 0.5f * w, 0.0f), IMGF);
  float y1 = fminf(fmaxf(cy - 0.5f * h, 0.0f), IMGF);
  float x2 = fminf(fmaxf(cx + 0.5f * w, 0.0f), IMGF);
  float y2 = fminf(fmaxf(cy + 0.5f * h, 0.0f), IMGF);
  ((float4*)boxes)[a] = make_float4(x1, y1, x2, y2);
}

// =====================================================================
// Kernel 1b: transpose logits (A,90) -> (90,A) through LDS tiles.
// Full tiles: ONE Tensor-Data-Mover DMA per workgroup (TENSORcnt);
// tail tile / no-TDM toolchain: per-lane async global->LDS (ASYNCcnt).
// Makes all 5 top-k selection passes fully coalesced afterwards.
// =====================================================================
__global__ __launch_bounds__(256) void transpose_kernel(
    const float* __restrict__ cls, float* __restrict__ clsT) {
  __shared__ __align__(16) float tile[TR_ANCH * NCLS];  // 45 KB
  const int t = threadIdx.x;
  const int a0 = blockIdx.x * TR_ANCH;
  const int na = min(TR_ANCH, A_TOT - a0);
  const size_t gbase = (size_t)a0 * NCLS;
  const unsigned ldsBase = (unsigned)(size_t)(&tile[0]);

  if (na == TR_ANCH) {
#if HAVE_TDM
    // ---- single TDM descriptor: 1D tile of 11520 x 4B, linear copy ----
    unsigned wid = __builtin_amdgcn_readfirstlane((unsigned)t >> 5);
    if (wid == 0) {  // uniform scalar branch: only wave 0 issues the DMA
      const unsigned long long ga =
          (unsigned long long)(cls + gbase);        // byte address
      const unsigned CNT = TR_ANCH * NCLS;          // 11520 elements
      v4u g0;
      g0.x = 0x1u;                                  // count=1 valid D#
      g0.y = ldsBase;                               // lds_addr [63:32]
      g0.z = (unsigned)(ga & 0xFFFFFFFFull);        // global_addr[31:0]
      g0.w = (unsigned)((ga >> 32) & 0x01FFFFFFull) // global_addr[56:32]
             | 0x80000000u;                         // type=2 ("image")
      v8i g1;
      g1[0] = (int)(2u << 16);                      // data_size=4B, mask=0
      g1[1] = (int)((CNT & 0xFFFFu) << 16);         // tensor_dim0 lo16
      g1[2] = (int)(((CNT >> 16) & 0xFFFFu) |       // tensor_dim0 hi16
                    (1u << 16));                    // tensor_dim1 = 1
      g1[3] = (int)((0u) | ((CNT & 0xFFFFu) << 16));// tile_dim0 = 11520
      g1[4] = (int)1u;                              // tile_dim1=1, tile_dim2=0
      g1[5] = (int)CNT;                             // dim0_stride lo32
      g1[6] = 0;                                    // dim0_stride hi16, dim1_stride lo16
      g1[7] = 0;                                    // dim1_stride hi32
      v4i gz = {0, 0, 0, 0};                        // groups 2/3: unused (<=2D)
#if __clang_major__ >= 23
      v8i gz8 = {0, 0, 0, 0, 0, 0, 0, 0};
      __builtin_amdgcn_tensor_load_to_lds(g0, g1, gz, gz, gz8, 0);
#else
      __builtin_amdgcn_tensor_load_to_lds(g0, g1, gz, gz, 0);
#endif
      __builtin_amdgcn_s_wait_tensorcnt(0);
    }
#else
    // full tile: 2880 x 16B async copies (linear)
    for (int k = t; k < (TR_ANCH * NCLS) / 4; k += 256) {
      unsigned lo = ldsBase + (unsigned)(k * 16);
      unsigned long long ga = (unsigned long long)(cls + gbase + (size_t)k * 4);
      asm volatile("global_load_async_to_lds_b128 %0, %1, off"
                   :: "v"(lo), "v"(ga) : "memory");
    }
    asm volatile("s_wait_asynccnt 0" ::: "memory");
#endif
  } else {
    // tail tile (20 anchors): 4B async copies
    const int cnt = na * NCLS;
    for (int k = t; k < cnt; k += 256) {
      unsigned lo = ldsBase + (unsigned)(k * 4);
      unsigned long long ga = (unsigned long long)(cls + gbase + (size_t)k);
      asm volatile("global_load_async_to_lds_b32 %0, %1, off"
                   :: "v"(lo), "v"(ga) : "memory");
    }
    asm volatile("s_wait_asynccnt 0" ::: "memory");
  }
  __syncthreads();

  if (na == TR_ANCH) {
    for (int k = t; k < NCLS * TR_ANCH; k += 256) {
      int c = k >> 7, a = k & (TR_ANCH - 1);  // TR_ANCH = 128
      clsT[(size_t)c * A_TOT + a0 + a] = tile[a * NCLS + c];
    }
  } else {
    const int cnt = NCLS * na;
    for (int k = t; k < cnt; k += 256) {
      int c = k / na, a = k % na;
      clsT[(size_t)c * A_TOT + a0 + a] = tile[a * NCLS + c];
    }
  }
}

// =====================================================================
// Kernel 2: per-class top-512 (radix select + bitonic sort).
// Element (c,a) at src[c*classPitch + a*step]:
//   transposed: classPitch=A_TOT, step=1  (coalesced)
//   fallback:   classPitch=1,    step=NCLS (strided, L2-resident)
// =====================================================================
__global__ __launch_bounds__(1024) void topk_kernel(
    const float* __restrict__ src, long long classPitch, int step,
    float* __restrict__ topVals, int* __restrict__ topIdx) {
  const int c = blockIdx.x;
  const int tid = threadIdx.x;
  const int NT = 1024;
  const float* __restrict__ p = src + (size_t)c * (size_t)classPitch;

  __shared__ unsigned hist[256];
  __shared__ unsigned long long skey[KC];
  __shared__ unsigned ties[1024];
  __shared__ unsigned scal[4];  // 0=bucket 1=cumAbove 2=cntGT 3=tieCnt

  // ---- 4-pass MSB-first radix select for the top-KC threshold ----
  unsigned prefix = 0, pmask = 0;
  int kneed = KC;
  for (int pass = 0; pass < 4; ++pass) {
    const int shift = 24 - 8 * pass;
    if (tid < 256) hist[tid] = 0;
    __syncthreads();
    for (int a = tid; a < A_TOT; a += NT) {
      if (pass == 0)  // CDNA5 global_prefetch_b8: warm the next chunk
        __builtin_prefetch(p + (size_t)(a + NT) * step, 0, 0);
      unsigned key = fkey(p[(size_t)a * step]);
      if ((key & pmask) == prefix)
        atomicAdd(&hist[(key >> shift) & 255u], 1u);
    }
    __syncthreads();
    if (tid == 0) {
      unsigned cum = 0;
      int b = 255;
#pragma unroll 1
      for (; b > 0; --b) {
        unsigned h = hist[b];
        if (cum + h >= (unsigned)kneed) break;
        cum += h;
      }
      scal[0] = (unsigned)b;
      scal[1] = cum;
    }
    __syncthreads();
    prefix |= scal[0] << shift;
    pmask  |= 0xFFu << shift;
    kneed  -= (int)scal[1];
    __syncthreads();
  }
  const unsigned T = prefix;  // threshold key; kneed elems == T still needed

  // ---- compact: strictly-greater elements + tie candidates ----
  if (tid == 0) { scal[2] = 0; scal[3] = 0; }
  __syncthreads();
  for (int a = tid; a < A_TOT; a += NT) {
    unsigned key = fkey(p[(size_t)a * step]);
    if (key > T) {
      unsigned pos = atomicAdd(&scal[2], 1u);
      skey[pos] = ((unsigned long long)key << 32) | (unsigned)(~a);
    } else if (key == T) {
      unsigned pos = atomicAdd(&scal[3], 1u);
      if (pos < 1024u) ties[pos] = (unsigned)a;
    }
  }
  __syncthreads();
  const unsigned cntGT = scal[2];
  const unsigned tcnt = scal[3] < 1024u ? scal[3] : 1024u;
  if (tid >= (int)tcnt) ties[tid] = 0xFFFFFFFFu;  // pad for sort
  __syncthreads();

  // ---- bitonic sort ties ascending by anchor index (lowest-index ties) ----
  for (unsigned k2 = 2; k2 <= 1024; k2 <<= 1) {
    for (unsigned jj = k2 >> 1; jj > 0; jj >>= 1) {
      unsigned i = (unsigned)tid, ixj = i ^ jj;
      if (ixj > i) {
        unsigned av = ties[i], bv = ties[ixj];
        bool asc = ((i & k2) == 0);
        if (asc ? (av > bv) : (av < bv)) { ties[i] = bv; ties[ixj] = av; }
      }
      __syncthreads();
    }
  }

  // ---- fill remaining slots with the smallest-index tie elements ----
  const int kneed2 = KC - (int)cntGT;
  for (int t = tid; t < kneed2; t += NT)
    skey[cntGT + t] = ((unsigned long long)T << 32) | (unsigned)(~ties[t]);
  __syncthreads();

  // ---- bitonic sort 512 (key,~idx) descending: jax.top_k order ----
  for (unsigned k2 = 2; k2 <= KC; k2 <<= 1) {
    for (unsigned jj = k2 >> 1; jj > 0; jj >>= 1) {
      if (tid < KC) {
        unsigned i = (unsigned)tid, ixj = i ^ jj;
        if (ixj > i) {
          unsigned long long av = skey[i], bv = skey[ixj];
          bool desc = ((i & k2) == 0);
          if (desc ? (av < bv) : (av > bv)) { skey[i] = bv; skey[ixj] = av; }
        }
      }
      __syncthreads();
    }
  }

  // ---- emit: sigmoid only on the 512 survivors ----
  if (tid < KC) {
    unsigned long long e = skey[tid];
    unsigned key = (unsigned)(e >> 32);
    int idx = (int)(~(unsigned)(e & 0xFFFFFFFFull));
    float logit = unfkey(key);
    topVals[c * KC + tid] = 1.0f / (1.0f + expf(-logit));
    topIdx[c * KC + tid] = idx;
  }
}

// =====================================================================
// Kernel 3: per-class greedy NMS (exact reference scan semantics).
// Candidate boxes gathered into LDS via async global->LDS (ASYNCcnt).
// =====================================================================
__global__ __launch_bounds__(KC) void nms_kernel(
    const float* __restrict__ boxes, const float* __restrict__ topVals,
    const int* __restrict__ topIdx, float* __restrict__ keptSc,
    float* __restrict__ candB) {
  const int c = blockIdx.x;
  const int j = threadIdx.x;
  __shared__ __align__(16) float4 sBox[KC];
  __shared__ int sKeep;

  int idx = topIdx[c * KC + j];
  float v = topVals[c * KC + j];

  // per-lane indexed gather: 16B per lane, global -> LDS, async path
  unsigned ldsOff = (unsigned)(size_t)(&sBox[j]);  // low 32b of generic = LDS offset
  unsigned long long gaddr = (unsigned long long)(boxes + (size_t)idx * 4);
  asm volatile("global_load_async_to_lds_b128 %0, %1, off"
               :: "v"(ldsOff), "v"(gaddr) : "memory");
  asm volatile("s_wait_asynccnt 0" ::: "memory");
  __syncthreads();

  float4 bj = sBox[j];
  ((float4*)candB)[c * KC + j] = bj;  // for final gather
  float areaj = (bj.z - bj.x) * (bj.w - bj.y);
  bool supp = false;
  const bool valid = v > THRESH;
  float myScore = -1.0f;

  for (int i = 0; i < KC; ++i) {
    if (j == i) sKeep = (valid && !supp) ? 1 : 0;
    __syncthreads();
    if (sKeep) {
      if (j == i) myScore = v;  // row i kept: record its score
      float4 bi = sBox[i];
      float xx1 = fmaxf(bi.x, bj.x), yy1 = fmaxf(bi.y, bj.y);
      float xx2 = fminf(bi.z, bj.z), yy2 = fminf(bi.w, bj.w);
      float inter = fmaxf(xx2 - xx1, 0.0f) * fmaxf(yy2 - yy1, 0.0f);
      float areai = (bi.z - bi.x) * (bi.w - bi.y);
      float iou = inter / (areai + areaj - inter + 1e-8f);
      if (iou > IOU_THR) supp = true;  // includes self (iou==1), as in ref
    }
    __syncthreads();
  }
  keptSc[c * KC + j] = myScore;  // kept ? val : -1
}

// =====================================================================
// Kernel 4: global top-100 over 46080 kept scores; write (100,7) output.
// Scores live in registers (45/thread, exact tiling); 100 argmax rounds.
// =====================================================================
__global__ __launch_bounds__(TPB_SEL) void topdet_kernel(
    const float* __restrict__ ks, const float* __restrict__ candB,
    float* __restrict__ out) {
  const int tid = threadIdx.x;
  __shared__ float rv[TPB_SEL];
  __shared__ int ri[TPB_SEL];

  float lv[EPT];
#pragma unroll
  for (int t = 0; t < EPT; ++t) lv[t] = ks[tid + (t << 10)];

  for (int r = 0; r < MAXDET; ++r) {
    float bv = -1e30f;
    int bi = 0;
#pragma unroll
    for (int t = 0; t < EPT; ++t) {  // ascending g => first win = lowest idx tie
      int g = tid + (t << 10);
      if (lv[t] > bv) { bv = lv[t]; bi = g; }
    }
    rv[tid] = bv; ri[tid] = bi;
    __syncthreads();
    for (int s = TPB_SEL >> 1; s > 0; s >>= 1) {
      if (tid < s) {
        float ov = rv[tid + s]; int oi = ri[tid + s];
        if (ov > rv[tid] || (ov == rv[tid] && oi < ri[tid])) {
          rv[tid] = ov; ri[tid] = oi;
        }
      }
      __syncthreads();
    }
    float best = rv[0];
    int bidx = ri[0];
    // consume winner (constant-index predicated update keeps lv in VGPRs)
    if ((bidx & 1023) == tid) {
      int tsel = bidx >> 10;
#pragma unroll
      for (int t = 0; t < EPT; ++t)
        if (t == tsel) lv[t] = -1e30f;
    }
    if (tid == 0) {
      bool ok = best > 0.0f;
      float x1 = 0, y1 = 0, x2 = 0, y2 = 0, sc = 0, lb = 0;
      if (ok) {
        const float* b = candB + (size_t)bidx * 4;
        x1 = b[0]; y1 = b[1]; x2 = b[2]; y2 = b[3];
        sc = best; lb = (float)(bidx / KC);
      }
      float* o = out + r * 7;
      o[0] = 0.0f; o[1] = x1; o[2] = y1; o[3] = x2; o[4] = y2;
      o[5] = sc; o[6] = lb;
    }
    __syncthreads();
  }
}

// =====================================================================
extern "C" void kernel_launch(void* const* d_in, const int* in_sizes, int n_in,
                              void* d_out, int out_size, void* d_ws,
                              size_t ws_size, hipStream_t stream) {
  (void)in_sizes; (void)n_in; (void)out_size;
  // setup_inputs order: x(unused), regression, classification, anchors
  const float* reg = (const float*)d_in[1];
  const float* cls = (const float*)d_in[2];
  const float* anc = (const float*)d_in[3];
  float* out = (float*)d_out;

  // workspace layout (all 16B aligned)
  float* boxes   = (float*)d_ws;               // A_TOT*4
  float* topVals = boxes + (size_t)A_TOT * 4;  // NFLAT
  int*   topIdx  = (int*)(topVals + NFLAT);    // NFLAT
  float* keptSc  = (float*)(topIdx + NFLAT);   // NFLAT
  float* candB   = keptSc + NFLAT;             // NFLAT*4
  float* clsT    = candB + (size_t)NFLAT * 4;  // A_TOT*NCLS (39.8 MB)

  const size_t baseBytes = (size_t)(candB + (size_t)NFLAT * 4 - (float*)d_ws) * 4;
  const size_t needT = baseBytes + (size_t)A_TOT * NCLS * 4;
  const bool useT = ws_size >= needT;  // fixed per run -> deterministic

  decode_kernel<<<(A_TOT + 255) / 256, 256, 0, stream>>>(reg, anc, boxes);
  if (useT) {
    const int nblk = (A_TOT + TR_ANCH - 1) / TR_ANCH;  // 864
    transpose_kernel<<<nblk, 256, 0, stream>>>(cls, clsT);
    topk_kernel<<<NCLS, 1024, 0, stream>>>(clsT, (long long)A_TOT, 1,
                                           topVals, topIdx);
  } else {
    topk_kernel<<<NCLS, 1024, 0, stream>>>(cls, 1LL, NCLS, topVals, topIdx);
  }
  nms_kernel<<<NCLS, KC, 0, stream>>>(boxes, topVals, topIdx, keptSc, candB);
  topdet_kernel<<<1, TPB_SEL, 0, stream>>>(keptSc, candB, out);
}